// TreeLSTMEncoder_78872779423848
// MI455X (gfx1250) — compile-verified
//
#include <hip/hip_runtime.h>

// ---------------------------------------------------------------------------
// TreeLSTM encoder for MI455X (gfx1250, wave32).
// fp32 end-to-end using V_WMMA_F32_16X16X4_F32 (exact fp32 accumulate,
// matches the fp32 reference numerics while using the matrix pipe).
// One wave (32 threads) owns one 16x16 output tile.
// ---------------------------------------------------------------------------

typedef __attribute__((ext_vector_type(2))) float v2f;
typedef __attribute__((ext_vector_type(8))) float v8f;

constexpr int B       = 8;
constexpr int NNODES  = 4095;
constexpr int F       = 256;
constexpr int H       = 256;
constexpr int DEPTH   = 12;

__device__ __forceinline__ v8f wmma_f32(v2f a, v2f b, v8f c) {
  // 8 args: (neg_a, A, neg_b, B, c_mod, C, reuse_a, reuse_b)
  return __builtin_amdgcn_wmma_f32_16x16x4_f32(
      false, a, false, b, (short)0, c, false, false);
}

__device__ __forceinline__ float sigmoidf_(float x) {
  return __builtin_amdgcn_rcpf(1.0f + __expf(-x));
}
__device__ __forceinline__ float tanhf_(float x) {
  // tanh(x) = 2*sigmoid(2x) - 1
  return 2.0f * __builtin_amdgcn_rcpf(1.0f + __expf(-2.0f * x)) - 1.0f;
}

// ---------------------------------------------------------------------------
// h0 = x @ W_e + b_e      (M = B*NNODES rows, K = F, N = H)
// grid.x = M tiles, grid.y = H/16, block = 32 (one wave)
// ---------------------------------------------------------------------------
__global__ __launch_bounds__(32) void embed_kernel(
    const float* __restrict__ x, const float* __restrict__ We,
    const float* __restrict__ be, float* __restrict__ h0, int Mrows) {
  const int lane = threadIdx.x & 31;
  const int m    = lane & 15;          // row within tile (A frag / store map)
  const int n    = lane & 15;          // col within tile (B frag / store map)
  const int koff = (lane >> 4) << 1;   // 0 or 2: which K pair this half-wave holds
  const int row0 = blockIdx.x * 16;
  const int ncol = blockIdx.y * 16;

  int rowA = row0 + m;
  float amask = (rowA < Mrows) ? 1.0f : 0.0f;
  int rowC = (rowA < Mrows) ? rowA : (Mrows - 1);

  const float* ax  = x  + (size_t)rowC * F + koff;
  const float* bw  = We + (size_t)koff * H + ncol + n;

  v8f acc = {};
  for (int k = 0; k < F; k += 4) {
    v2f a = *(const v2f*)(ax + k);      // global_load_b64 (8B aligned)
    a.x *= amask; a.y *= amask;
    v2f bb;
    bb.x = bw[(size_t)k * H];
    bb.y = bw[(size_t)k * H + H];
    acc = wmma_f32(a, bb, acc);
  }

  const float bias  = be[ncol + n];
  const int   mbase = (lane >> 4) << 3;   // 0 or 8 (C/D layout)
  for (int v = 0; v < 8; ++v) {
    int rm = row0 + mbase + v;
    if (rm < Mrows) h0[(size_t)rm * H + ncol + n] = acc[v] + bias;
  }
}

// ---------------------------------------------------------------------------
// One tree level.  Rows r in [0, B*2^l): b = r >> l, j = r & (2^l - 1).
// Computes for a 16x16 (rows x hidden-col) tile the six GEMM accumulators
// i, o, u, xf, hl@Uf, hr@Uf, then the gate math, writing h and c.
// ---------------------------------------------------------------------------
__global__ __launch_bounds__(32) void level_kernel(
    const float* __restrict__ h0, float* __restrict__ h, float* __restrict__ c,
    const float* __restrict__ Wiou, const float* __restrict__ Uiou,
    const float* __restrict__ biou, const float* __restrict__ Wf,
    const float* __restrict__ Uf,   const float* __restrict__ bfv,
    int level) {
  const int  nn    = 1 << level;
  const int  base  = nn - 1;
  const int  Mrows = B * nn;
  const bool leaf  = (level == DEPTH - 1);

  const int lane = threadIdx.x & 31;
  const int m    = lane & 15;
  const int n    = lane & 15;
  const int koff = (lane >> 4) << 1;
  const int row0 = blockIdx.x * 16;
  const int ncol = blockIdx.y * 16;

  int rowA   = row0 + m;
  float amask = (rowA < Mrows) ? 1.0f : 0.0f;
  int rowC   = (rowA < Mrows) ? rowA : (Mrows - 1);
  int jA     = rowC & (nn - 1);
  int bA     = rowC >> level;
  int nodeA  = base + jA;

  const float* xrow = h0 + ((size_t)bA * NNODES + nodeA) * H + koff;
  const float* hlrow = xrow;   // dummy init; only used when !leaf
  const float* hrrow = xrow;
  if (!leaf) {
    int lch = 2 * nodeA + 1, rch = 2 * nodeA + 2;
    hlrow = h + ((size_t)bA * NNODES + lch) * H + koff;
    hrrow = h + ((size_t)bA * NNODES + rch) * H + koff;
  }

  const int ld3 = 3 * H;
  const float* pWi = Wiou + (size_t)koff * ld3 + ncol + n;           // i block
  const float* pWo = pWi + H;                                        // o block
  const float* pWu = pWi + 2 * H;                                    // u block
  const float* pWf = Wf   + (size_t)koff * H   + ncol + n;
  const float* pUi = Uiou + (size_t)koff * ld3 + ncol + n;
  const float* pUo = pUi + H;
  const float* pUu = pUi + 2 * H;
  const float* pUf = Uf   + (size_t)koff * H   + ncol + n;

  v8f acc_i = {}, acc_o = {}, acc_u = {}, acc_xf = {}, acc_fl = {}, acc_fr = {};

  // Phase 1: A = xn  ->  i,o,u (W_iou) and xf (W_f)
  for (int k = 0; k < H; k += 4) {
    v2f a = *(const v2f*)(xrow + k);
    a.x *= amask; a.y *= amask;
    size_t o3 = (size_t)k * ld3;
    size_t o1 = (size_t)k * H;
    v2f bi; bi.x = pWi[o3]; bi.y = pWi[o3 + ld3];
    v2f bo; bo.x = pWo[o3]; bo.y = pWo[o3 + ld3];
    v2f bu; bu.x = pWu[o3]; bu.y = pWu[o3 + ld3];
    v2f bw; bw.x = pWf[o1]; bw.y = pWf[o1 + H];
    acc_i  = wmma_f32(a, bi, acc_i);
    acc_o  = wmma_f32(a, bo, acc_o);
    acc_u  = wmma_f32(a, bu, acc_u);
    acc_xf = wmma_f32(a, bw, acc_xf);
  }

  // Phase 2: A = hl, hr, hsum  ->  i,o,u (U_iou), fl/fr (U_f).  Uniform skip at leaf.
  if (!leaf) {
    for (int k = 0; k < H; k += 4) {
      v2f al = *(const v2f*)(hlrow + k);
      v2f ar = *(const v2f*)(hrrow + k);
      al.x *= amask; al.y *= amask;
      ar.x *= amask; ar.y *= amask;
      v2f as; as.x = al.x + ar.x; as.y = al.y + ar.y;
      size_t o3 = (size_t)k * ld3;
      size_t o1 = (size_t)k * H;
      v2f bi; bi.x = pUi[o3]; bi.y = pUi[o3 + ld3];
      v2f bo; bo.x = pUo[o3]; bo.y = pUo[o3 + ld3];
      v2f bu; bu.x = pUu[o3]; bu.y = pUu[o3 + ld3];
      v2f bu_f; bu_f.x = pUf[o1]; bu_f.y = pUf[o1 + H];
      acc_i  = wmma_f32(as, bi,   acc_i);
      acc_o  = wmma_f32(as, bo,   acc_o);
      acc_u  = wmma_f32(as, bu,   acc_u);
      acc_fl = wmma_f32(al, bu_f, acc_fl);
      acc_fr = wmma_f32(ar, bu_f, acc_fr);
    }
  }

  // Gate math + writeback (C/D layout: vgpr v, half-wave -> m = mbase + v)
  const float bi_ = biou[ncol + n];
  const float bo_ = biou[H + ncol + n];
  const float bu_ = biou[2 * H + ncol + n];
  const float bf_ = bfv[ncol + n];
  const int mbase = (lane >> 4) << 3;

  for (int v = 0; v < 8; ++v) {
    int rm = row0 + mbase + v;
    if (rm >= Mrows) continue;
    int j = rm & (nn - 1);
    int b = rm >> level;
    int node = base + j;
    size_t outoff = ((size_t)b * NNODES + node) * H + ncol + n;

    float iv  = sigmoidf_(acc_i[v] + bi_);
    float ov  = sigmoidf_(acc_o[v] + bo_);
    float uv  = tanhf_(acc_u[v] + bu_);
    float xfv = acc_xf[v] + bf_;

    float cn;
    if (!leaf) {
      int lch = 2 * node + 1, rch = 2 * node + 2;
      float clv = c[((size_t)b * NNODES + lch) * H + ncol + n];
      float crv = c[((size_t)b * NNODES + rch) * H + ncol + n];
      float flv = sigmoidf_(xfv + acc_fl[v]);
      float frv = sigmoidf_(xfv + acc_fr[v]);
      cn = iv * uv + flv * clv + frv * crv;
    } else {
      cn = iv * uv;   // cl = cr = 0 at the leaf level
    }
    float hv = ov * tanhf_(cn);
    c[outoff] = cn;
    h[outoff] = hv;
  }
}

// ---------------------------------------------------------------------------
// out = h[:, 0, :]   (B x H)
// ---------------------------------------------------------------------------
__global__ void copyout_kernel(const float* __restrict__ h, float* __restrict__ out) {
  int t = blockIdx.x * blockDim.x + threadIdx.x;
  if (t < B * H) {
    int b = t / H, col = t % H;
    out[t] = h[(size_t)b * NNODES * H + col];
  }
}

// ---------------------------------------------------------------------------
extern "C" void kernel_launch(void* const* d_in, const int* in_sizes, int n_in,
                              void* d_out, int out_size, void* d_ws, size_t ws_size,
                              hipStream_t stream) {
  const float* x    = (const float*)d_in[0];
  const float* We   = (const float*)d_in[1];
  const float* be   = (const float*)d_in[2];
  const float* Wiou = (const float*)d_in[3];
  const float* Uiou = (const float*)d_in[4];
  const float* biou = (const float*)d_in[5];
  const float* Wf   = (const float*)d_in[6];
  const float* Uf   = (const float*)d_in[7];
  const float* bfv  = (const float*)d_in[8];

  const size_t planes = (size_t)B * NNODES * H;   // 8,386,560 floats
  float* h0 = (float*)d_ws;
  float* h  = h0 + planes;
  float* c  = h  + planes;

  // 1) Embedding GEMM: h0 = x @ W_e + b_e
  {
    int Mrows = B * NNODES;
    dim3 grid((Mrows + 15) / 16, H / 16);
    embed_kernel<<<grid, 32, 0, stream>>>(x, We, be, h0, Mrows);
  }

  // 2) Bottom-up tree levels (serial dependency across levels)
  for (int l = DEPTH - 1; l >= 0; --l) {
    int Mrows = B * (1 << l);
    dim3 grid((Mrows + 15) / 16, H / 16);
    level_kernel<<<grid, 32, 0, stream>>>(h0, h, c, Wiou, Uiou, biou, Wf, Uf, bfv, l);
  }

  // 3) Emit h[:, 0, :]
  copyout_kernel<<<(B * H + 255) / 256, 256, 0, stream>>>(h, (float*)d_out);
}